// TopkRouting_49289044688910
// MI455X (gfx1250) — compile-verified
//
#include <hip/hip_runtime.h>

// TopkRouting for MI455X (gfx1250, wave32, WMMA).
//   Q,K: (1024, 196, 256) f32.  logits = (Q * 256^-0.5) @ K^T  per batch.
//   top-4 per row -> softmax(4).  Outputs: weights then indices, flat.

#define NB    1024
#define P2    196
#define CDIM  256
#define PADP  208          // 13 * 16
#define NCT   13           // col tiles of 16
#define NTOPK 4
#define SCALE 0.0625f      // 256^-0.5

typedef __attribute__((ext_vector_type(2))) float fx2;
typedef __attribute__((ext_vector_type(8))) float fx8;

__global__ __launch_bounds__(256)
void topk_routing_kernel(const float* __restrict__ Q,
                         const float* __restrict__ K,
                         float* __restrict__ out_w,
                         float* __restrict__ out_i) {
    __shared__ float lgt[16 * PADP];            // 16 rows x 208 padded cols
    __shared__ float pv[16 * 16 * NTOPK];       // partial top-4 values
    __shared__ int   pi[16 * 16 * NTOPK];       // partial top-4 indices

    const int n    = blockIdx.y;
    const int rt   = blockIdx.x;                // row tile 0..12
    const int tid  = threadIdx.x;
    const int wave = tid >> 5;                  // wave32
    const int lane = tid & 31;
    const int g    = lane >> 4;                 // half-wave group (K select)
    const int r    = lane & 15;                 // row/col within tile

    // ---- Phase 1: logits via V_WMMA_F32_16X16X4_F32 ----
    // A-frag: lane holds Q row (rt*16+r), K = {4s+2g, 4s+2g+1}
    // B-frag: lane holds K row (ct*16+r), same K pattern (B = K^T)
    const int qrow = min(rt * 16 + r, P2 - 1);
    const float* qbase = Q + ((size_t)n * P2 + qrow) * CDIM + 2 * g;

    for (int ct = wave; ct < NCT; ct += 8) {    // wave-uniform: EXEC stays full
        const int krow = min(ct * 16 + r, P2 - 1);
        const float* kbase = K + ((size_t)n * P2 + krow) * CDIM + 2 * g;

        fx8 acc = {};
#pragma unroll 8
        for (int s = 0; s < CDIM / 4; ++s) {
            fx2 a = *(const fx2*)(qbase + 4 * s);
            fx2 b = *(const fx2*)(kbase + 4 * s);
            acc = __builtin_amdgcn_wmma_f32_16x16x4_f32(
                false, a, false, b, (short)0, acc, false, false);
        }

        // C/D layout: lane holds col N = r (of this tile), rows M = i + 8*g
        const int   col   = ct * 16 + r;
        const float fill  = -__builtin_inff();
        const bool  colok = (col < P2);
#pragma unroll
        for (int i = 0; i < 8; ++i) {
            const int m = i + 8 * g;
            lgt[m * PADP + col] = colok ? acc[i] * SCALE : fill;
        }
    }
    __syncthreads();

    // ---- Phase 2a: per-(row, segment) partial top-4 over 13 cols ----
    {
        const int row = tid & 15;
        const int seg = tid >> 4;               // 0..15, 16 cols-groups of 13
        float v0 = -__builtin_inff(), v1 = v0, v2 = v0, v3 = v0;
        int   i0 = -1, i1 = -1, i2 = -1, i3 = -1;
        const int j0 = seg * 13;
#pragma unroll
        for (int j = j0; j < j0 + 13; ++j) {
            const float x = lgt[row * PADP + j];
            if (x > v3) {
                if (x > v0)      { v3=v2;i3=i2; v2=v1;i2=i1; v1=v0;i1=i0; v0=x;i0=j; }
                else if (x > v1) { v3=v2;i3=i2; v2=v1;i2=i1; v1=x;i1=j; }
                else if (x > v2) { v3=v2;i3=i2; v2=x;i2=j; }
                else             { v3=x;i3=j; }
            }
        }
        const int base = (row * 16 + seg) * NTOPK;
        pv[base+0]=v0; pv[base+1]=v1; pv[base+2]=v2; pv[base+3]=v3;
        pi[base+0]=i0; pi[base+1]=i1; pi[base+2]=i2; pi[base+3]=i3;
    }
    __syncthreads();

    // ---- Phase 2b: merge 16 partials per row, softmax(4), write out ----
    if (tid < 16) {
        const int row  = tid;
        const int grow = rt * 16 + row;
        if (grow < P2) {
            float v0 = -__builtin_inff(), v1 = v0, v2 = v0, v3 = v0;
            int   i0 = -1, i1 = -1, i2 = -1, i3 = -1;
            // scan seg-ascending, rank-ascending: strict '>' keeps the
            // lowest index among equal values (jax top_k tie-break)
            for (int seg = 0; seg < 16; ++seg) {
#pragma unroll
                for (int t = 0; t < NTOPK; ++t) {
                    const int   base = (row * 16 + seg) * NTOPK + t;
                    const float x  = pv[base];
                    const int   ix = pi[base];
                    if (x > v3) {
                        if (x > v0)      { v3=v2;i3=i2; v2=v1;i2=i1; v1=v0;i1=i0; v0=x;i0=ix; }
                        else if (x > v1) { v3=v2;i3=i2; v2=v1;i2=i1; v1=x;i1=ix; }
                        else if (x > v2) { v3=v2;i3=i2; v2=x;i2=ix; }
                        else             { v3=x;i3=ix; }
                    }
                }
            }
            // softmax over the 4 selected logits (v0 is the max)
            const float e0 = 1.0f;
            const float e1 = __expf(v1 - v0);
            const float e2 = __expf(v2 - v0);
            const float e3 = __expf(v3 - v0);
            const float inv = 1.0f / (e0 + e1 + e2 + e3);

            const size_t base = ((size_t)n * P2 + grow) * NTOPK;
            out_w[base + 0] = e0 * inv;
            out_w[base + 1] = e1 * inv;
            out_w[base + 2] = e2 * inv;
            out_w[base + 3] = e3 * inv;
            out_i[base + 0] = (float)i0;
            out_i[base + 1] = (float)i1;
            out_i[base + 2] = (float)i2;
            out_i[base + 3] = (float)i3;
        }
    }
}

extern "C" void kernel_launch(void* const* d_in, const int* in_sizes, int n_in,
                              void* d_out, int out_size, void* d_ws, size_t ws_size,
                              hipStream_t stream) {
    (void)in_sizes; (void)n_in; (void)d_ws; (void)ws_size; (void)out_size;
    const float* Q = (const float*)d_in[0];
    const float* K = (const float*)d_in[1];
    float* out_w = (float*)d_out;                              // (1024,196,4) weights
    float* out_i = out_w + (size_t)NB * P2 * NTOPK;            // (1024,196,4) indices

    dim3 grid(NCT, NB);   // 13 row tiles x 1024 batches
    dim3 block(256);      // 8 wave32
    topk_routing_kernel<<<grid, block, 0, stream>>>(Q, K, out_w, out_i);
}